// DeformConv3DWrapper_16277926052481
// MI455X (gfx1250) — compile-verified
//
#include <hip/hip_runtime.h>
#include <hip/hip_bf16.h>

// MI455X (gfx1250) deformable 3D conv forward.
// Pipeline: pad -> offset WMMA-GEMM (+bias) -> fused trilinear-gather + final WMMA-GEMM.
// All working tensors are L2-resident; compute path uses V_WMMA_F32_16X16X4_F32;
// the final-GEMM weight tiles are staged into LDS with async global->LDS copies
// (ASYNCcnt-tracked), overlapped with the data-dependent gather staging of A.

typedef __attribute__((ext_vector_type(2))) float v2f;
typedef __attribute__((ext_vector_type(8))) float v8f;

// pointer types for the gfx1250 async global->LDS copy builtins
typedef int v4i_ __attribute__((vector_size(16)));
typedef __attribute__((address_space(1))) v4i_* gas_v4i;   // global
typedef __attribute__((address_space(3))) v4i_* las_v4i;   // LDS

#define C_IN   64
#define T_D    8
#define H_D    32
#define W_D    32
#define NPOS   8192          // 8*32*32
#define KDIM   1728          // 64*27
#define NSTEP  432           // KDIM/4
#define XP_Z   10
#define XP_Y   34
#define XP_X   34
#define XP_CH  (XP_Z*XP_Y*XP_X)   // 11560
#define XP_N   (C_IN*XP_CH)       // 739840
#define NOFF   81
#define C_OUT  128

// K2 tiling
#define M_BLK  64            // positions per block (4 WMMA M-subtiles)
#define KC     216           // K-chunk: 8 input channels * 27 taps
#define NCHUNK 8             // KDIM / KC

// ---------------------------------------------------------------------------
// K0: zero-pad x (pad=1) -> xp[64][10][34][34]
// ---------------------------------------------------------------------------
__global__ void pad_kernel(const float* __restrict__ x, float* __restrict__ xp) {
  int idx = blockIdx.x * blockDim.x + threadIdx.x;
  if (idx >= XP_N) return;
  int c  = idx / XP_CH;
  int r  = idx - c * XP_CH;
  int z  = r / (XP_Y * XP_X);
  int r2 = r - z * (XP_Y * XP_X);
  int y  = r2 / XP_X;
  int xx = r2 - y * XP_X;
  float v = 0.0f;
  if (z >= 1 && z <= T_D && y >= 1 && y <= H_D && xx >= 1 && xx <= W_D)
    v = x[((c * T_D + (z - 1)) * H_D + (y - 1)) * W_D + (xx - 1)];
  xp[idx] = v;
}

// ---------------------------------------------------------------------------
// K1: offset conv as implicit GEMM.  offs[o][pos] = sum_k w_off[o][k]*A[pos][k] + b_off[o]
// A[pos][k] = xp[c][z+dz][y+dy][x+dx],  k = c*27 + (dz*9+dy*3+dx)
// grid (512 pos-tiles, 6 outch-tiles of 16; 96 >= 81), one wave per tile.
// ---------------------------------------------------------------------------
__global__ void __launch_bounds__(32) offset_gemm_kernel(
    const float* __restrict__ xp, const float* __restrict__ w_off,
    const float* __restrict__ b_off, float* __restrict__ offs) {
  const int lane = threadIdx.x & 31;
  const int mr   = lane & 15;          // A row (position) / D column index source
  const int half = lane >> 4;          // K sub-block: 0 -> K{0,1}, 1 -> K{2,3}
  const int pos  = blockIdx.x * 16 + mr;
  const int z    = pos >> 10;
  const int y    = (pos >> 5) & 31;
  const int xx   = pos & 31;
  const int o    = blockIdx.y * 16 + mr;
  const int oc   = (o < NOFF) ? o : (NOFF - 1);
  const float bsc = (o < NOFF) ? 1.0f : 0.0f;   // mask padded out-channels

  v8f acc = {0.f, 0.f, 0.f, 0.f, 0.f, 0.f, 0.f, 0.f};

  for (int s = 0; s < NSTEP; ++s) {
    const int k0 = 4 * s + 2 * half;
    v2f av, bv;
#pragma unroll
    for (int t = 0; t < 2; ++t) {
      const int k   = k0 + t;
      const int c   = k / 27;
      const int tap = k - 27 * c;
      const int dz  = tap / 9;
      const int r9  = tap - 9 * dz;
      const int dy  = r9 / 3;
      const int dx  = r9 - 3 * dy;
      av[t] = xp[((c * XP_Z + z + dz) * XP_Y + (y + dy)) * XP_X + (xx + dx)];
    }
    bv.x = w_off[oc * KDIM + k0] * bsc;
    bv.y = w_off[oc * KDIM + k0 + 1] * bsc;
    acc = __builtin_amdgcn_wmma_f32_16x16x4_f32(false, av, false, bv,
                                                (short)0, acc, false, false);
  }

  if (o < NOFF) {
    const float bias = b_off[o];
#pragma unroll
    for (int v = 0; v < 8; ++v)
      offs[o * NPOS + blockIdx.x * 16 + v + 8 * half] = acc[v] + bias;
  }
}

// ---------------------------------------------------------------------------
// K2: fused trilinear sampling + final GEMM, M_BLK=64 positions per block.
// Per K-chunk (8 input channels x 27 taps = 216):
//   - async-copy B chunk w_conv[128][216] into LDS (ASYNCcnt path),
//   - gather/trilinear-stage A chunk [216][64] into LDS,
//   - s_wait_asynccnt 0 + barrier, then 8 waves x 4 M-subtiles of f32 WMMA.
// Scrambled tap mapping (derived from _reshape_x_offset + stride-3 conv):
//   s1 = a*32 + J ; j' = s1/3 ; q = s1%3
//   s2 = q*96 + 3K + c ; k' = s2/9 ; m = s2%9 ; n = 9b + m
//   A[pos][ch*27 + tap] = trilinear sample n at source position (I, j', k').
// ---------------------------------------------------------------------------
__global__ void __launch_bounds__(256) deform_gemm_kernel(
    const float* __restrict__ xp, const float* __restrict__ offs,
    const float* __restrict__ w_conv, float* __restrict__ out) {
  __shared__ float Atile[(KC / 2) * 2 * M_BLK];  // [108][64][2] = 55296 B
  __shared__ float Btile[C_OUT * KC];            // [128][216]   = 110592 B

  const int tid  = threadIdx.x;
  const int p0   = blockIdx.x * M_BLK;
  const int lane = tid & 31;
  const int wv   = tid >> 5;           // 0..7 -> output-channel tile
  const int mr   = lane & 15;
  const int half = lane >> 4;
  const int o    = wv * 16 + mr;

  v8f acc[4];
#pragma unroll
  for (int mt = 0; mt < 4; ++mt)
    acc[mt] = (v8f){0.f, 0.f, 0.f, 0.f, 0.f, 0.f, 0.f, 0.f};

  for (int ch = 0; ch < NCHUNK; ++ch) {
    const int c0 = ch * 8;             // first input channel of this chunk

    // ---- async-stage B chunk: w_conv[o][ch*216 .. +216) -> Btile[o][*] ----
    for (int i = tid; i < (C_OUT * KC) / 4; i += 256) {
      const int f  = i * 4;            // float index into Btile
      const int oo = f / KC;
      const int kk = f - oo * KC;
      const float* gsrc = w_conv + oo * KDIM + ch * KC + kk;   // 16B aligned
#if __has_builtin(__builtin_amdgcn_global_load_async_to_lds_b128)
      __builtin_amdgcn_global_load_async_to_lds_b128(
          (gas_v4i)gsrc, (las_v4i)&Btile[f], 0, 0);
#else
      *(float4*)&Btile[f] = *(const float4*)gsrc;
#endif
    }

    // ---- gather-stage A chunk: 27 taps x 64 positions, 8 channels each ----
    for (int task = tid; task < 27 * M_BLK; task += 256) {
      const int tap = task >> 6;       // task = tap*64 + m
      const int m   = task & 63;
      const int a   = tap / 9;
      const int r9  = tap - 9 * a;
      const int bb  = r9 / 3;
      const int cc  = r9 - 3 * bb;

      const int pos = p0 + m;
      const int I   = pos >> 10;
      const int J   = (pos >> 5) & 31;
      const int K   = pos & 31;

      const int s1 = a * 32 + J;
      const int jp = s1 / 3;
      const int q  = s1 - 3 * jp;
      const int s2 = q * 96 + 3 * K + cc;
      const int kp = s2 / 9;
      const int mm = s2 - 9 * kp;
      const int n  = bb * 9 + mm;
      const int sp = I * 1024 + jp * 32 + kp;

      // deformed coordinates (reproduce reference's cross-axis normalization)
      const float pz = (float)(I  + (n / 9)       - 1) + offs[n * NPOS + sp];
      const float py = (float)(jp + ((n / 3) % 3) - 1) + offs[(27 + n) * NPOS + sp];
      const float px = (float)(kp + (n % 3)       - 1) + offs[(54 + n) * NPOS + sp];

      float ix = fminf(fmaxf(pz * (33.0f / 9.0f), 0.0f), 33.0f);  // W axis (34)
      float iy = fminf(fmaxf(py,                  0.0f), 33.0f);  // H axis (34)
      float iz = fminf(fmaxf(px * (9.0f / 33.0f), 0.0f), 9.0f);   // D axis (10)

      const float fx = floorf(ix), wx = ix - fx;
      const float fy = floorf(iy), wy = iy - fy;
      const float fz = floorf(iz), wz = iz - fz;
      const int x0 = (int)fx, x1 = (x0 + 1 < 33) ? x0 + 1 : 33;
      const int y0 = (int)fy, y1 = (y0 + 1 < 33) ? y0 + 1 : 33;
      const int z0 = (int)fz, z1 = (z0 + 1 < 9) ? z0 + 1 : 9;

      const int i000 = (z0 * XP_Y + y0) * XP_X + x0;
      const int i001 = (z0 * XP_Y + y0) * XP_X + x1;
      const int i010 = (z0 * XP_Y + y1) * XP_X + x0;
      const int i011 = (z0 * XP_Y + y1) * XP_X + x1;
      const int i100 = (z1 * XP_Y + y0) * XP_X + x0;
      const int i101 = (z1 * XP_Y + y0) * XP_X + x1;
      const int i110 = (z1 * XP_Y + y1) * XP_X + x0;
      const int i111 = (z1 * XP_Y + y1) * XP_X + x1;

      const float omwx = 1.0f - wx, omwy = 1.0f - wy, omwz = 1.0f - wz;

#pragma unroll
      for (int cl = 0; cl < 8; ++cl) {
        const float* b = xp + (c0 + cl) * XP_CH;
        const float g000 = b[i000], g001 = b[i001];
        const float g010 = b[i010], g011 = b[i011];
        const float g100 = b[i100], g101 = b[i101];
        const float g110 = b[i110], g111 = b[i111];
        const float c00 = g000 * omwx + g001 * wx;
        const float c01 = g010 * omwx + g011 * wx;
        const float c10 = g100 * omwx + g101 * wx;
        const float c11 = g110 * omwx + g111 * wx;
        const float cA  = c00 * omwy + c01 * wy;
        const float cB  = c10 * omwy + c11 * wy;
        const float val = cA * omwz + cB * wz;
        const int kdl = cl * 27 + tap;          // local K within chunk
        Atile[(kdl >> 1) * (2 * M_BLK) + m * 2 + (kdl & 1)] = val;
      }
    }

#if __has_builtin(__builtin_amdgcn_s_wait_asynccnt)
    __builtin_amdgcn_s_wait_asynccnt(0);
#endif
    __syncthreads();

    // ---- WMMA over chunk: 54 steps x 4 M-subtiles per wave ----
    for (int s = 0; s < KC / 4; ++s) {
      const int kp2 = 2 * s + half;    // K pair index: K = {2*kp2, 2*kp2+1}
      const v2f bv = *(const v2f*)&Btile[o * KC + kp2 * 2];
#pragma unroll
      for (int mt = 0; mt < 4; ++mt) {
        const v2f av = *(const v2f*)&Atile[kp2 * (2 * M_BLK) + (mt * 16 + mr) * 2];
        acc[mt] = __builtin_amdgcn_wmma_f32_16x16x4_f32(false, av, false, bv,
                                                        (short)0, acc[mt], false, false);
      }
    }
    __syncthreads();
  }

  // ---- epilogue: D element (M = mt*16 + v + 8*half, N = mr) ----
#pragma unroll
  for (int mt = 0; mt < 4; ++mt)
#pragma unroll
    for (int v = 0; v < 8; ++v)
      out[o * NPOS + p0 + mt * 16 + v + 8 * half] = acc[mt][v];
}

// ---------------------------------------------------------------------------
extern "C" void kernel_launch(void* const* d_in, const int* in_sizes, int n_in,
                              void* d_out, int out_size, void* d_ws, size_t ws_size,
                              hipStream_t stream) {
  const float* x      = (const float*)d_in[0];   // [1,64,8,32,32]
  const float* w_off  = (const float*)d_in[1];   // [81,64,3,3,3]
  const float* b_off  = (const float*)d_in[2];   // [81]
  const float* w_conv = (const float*)d_in[3];   // [128,64,3,3,3]
  float* out = (float*)d_out;                    // [1,128,8,32,32]

  float* xp   = (float*)d_ws;                    // 739840 floats (2.96 MB)
  float* offs = xp + XP_N;                       // 81*8192 floats (2.65 MB)

  pad_kernel<<<(XP_N + 255) / 256, 256, 0, stream>>>(x, xp);
  offset_gemm_kernel<<<dim3(NPOS / 16, 6), 32, 0, stream>>>(xp, w_off, b_off, offs);
  deform_gemm_kernel<<<NPOS / M_BLK, 256, 0, stream>>>(xp, offs, w_conv, out);
}